// FullyConnectedClassifier_42494406426970
// MI455X (gfx1250) — compile-verified
//
#include <hip/hip_runtime.h>

#define NUM_CAM 16
#define IN_SZ   256
#define OUT_SZ  128
#define BATCH   8192
#define TILE_M  16
#define XPITCH  260   // fp32 elements per staged X row (260 % 64 == 4 -> conflict-free b64 reads)
#define WPITCH  130   // float2 elements per k-pair row of transposed W

typedef __attribute__((ext_vector_type(2))) float v2f;
typedef __attribute__((ext_vector_type(8))) float v8f;

// ---------------------------------------------------------------------------
// Phase 0: zero per-camera counters in workspace
// ---------------------------------------------------------------------------
__global__ void fc_zero_counts(int* counts) {
    if (threadIdx.x < NUM_CAM) counts[threadIdx.x] = 0;
}

// ---------------------------------------------------------------------------
// Phase 1: bucket sample indices by camera (grouped-GEMM row gather lists)
// ---------------------------------------------------------------------------
__global__ void fc_bucket(const int* __restrict__ cam,
                          int* __restrict__ counts,
                          int* __restrict__ idx) {
    int t = blockIdx.x * blockDim.x + threadIdx.x;
    if (t < BATCH) {
        int c = cam[t] & (NUM_CAM - 1);
        int pos = atomicAdd(&counts[c], 1);
        idx[c * BATCH + pos] = t;
    }
}

// ---------------------------------------------------------------------------
// Phase 2: grouped GEMM, one (camera, 16-row tile) per block, 8 waves cover
// the 128 output channels with v_wmma_f32_16x16x4_f32 (fp32-exact).
// ---------------------------------------------------------------------------
__launch_bounds__(256)
__global__ void fc_expert_gemm(const float* __restrict__ x,
                               const float* __restrict__ W,
                               const float* __restrict__ bias,
                               const int*  __restrict__ counts,
                               const int*  __restrict__ idx,
                               float* __restrict__ out) {
    __shared__ float Xs[TILE_M * XPITCH];      // 16 gathered rows of x
    __shared__ v2f   Wt[128 * WPITCH];         // Wt[p][n] = {W[c][n][2p], W[c][n][2p+1]}
    __shared__ int   Si[TILE_M];               // gathered sample indices for epilogue

    const int c     = blockIdx.x;
    const int tile  = blockIdx.y;
    const int count = counts[c];
    if (tile * TILE_M >= count) return;        // uniform: whole block exits together

    const int tid  = threadIdx.x;
    const int lane = tid & 31;
    const int wave = tid >> 5;                 // 0..7
    const int half = lane >> 4;                // 0 | 1  (half-wave select per ISA layout)
    const int l    = lane & 15;                // 0..15

    // ---- stage gathered X rows: 16 threads/row, 16 floats/thread (4x float4)
    {
        int row  = tid >> 4;
        int col0 = (tid & 15) * 16;
        int r    = tile * TILE_M + row;
        int s    = idx[c * BATCH + min(r, count - 1)];      // clamp padded rows
        if ((tid & 15) == 0) Si[row] = s;
        const float4* src = (const float4*)(x + (size_t)s * IN_SZ + col0);
        float* dst = &Xs[row * XPITCH + col0];
        #pragma unroll
        for (int i = 0; i < 4; ++i) {
            float4 v = src[i];
            dst[i * 4 + 0] = v.x; dst[i * 4 + 1] = v.y;
            dst[i * 4 + 2] = v.z; dst[i * 4 + 3] = v.w;
        }
    }

    // ---- stage W[c] as k-pair-major float2: coalesced global reads, LDS scatter
    {
        const v2f* Wsrc = (const v2f*)(W + (size_t)c * OUT_SZ * IN_SZ);
        int p     = tid & 127;                 // k-pair index 0..127
        int nbase = tid >> 7;                  // 0 | 1
        #pragma unroll 4
        for (int it = 0; it < 64; ++it) {
            int n = it * 2 + nbase;            // output channel 0..127
            Wt[p * WPITCH + n] = Wsrc[n * 128 + p];
        }
    }
    __syncthreads();

    // ---- WMMA main loop: D(16x16) += A(16x4) x B(4x16), K = 256 in steps of 4
    // A layout (ISA 7.12.2, 32-bit 16x4): lanes 0-15 hold M=lane K={k,k+1},
    //                                     lanes 16-31 hold M=lane-16 K={k+2,k+3}
    // B mirror: lanes 0-15 hold N=lane K={k,k+1}, lanes 16-31 N=lane-16 K={k+2,k+3}
    // Outer x8 / inner unrolled x8 so every LDS access inside the body is a
    // compile-time offset (< 64KB) from a pointer bumped once per outer iter.
    const int n0 = wave * 16;
    const float* ap = &Xs[l * XPITCH + 2 * half];
    const v2f*   bp = &Wt[half * WPITCH + (n0 + l)];

    v8f acc = {};
    for (int ko = 0; ko < 8; ++ko) {
        #pragma unroll
        for (int ki = 0; ki < 8; ++ki) {
            v2f a = *(const v2f*)(ap + ki * 4);
            v2f b = bp[ki * 2 * WPITCH];
            acc = __builtin_amdgcn_wmma_f32_16x16x4_f32(
                /*neg_a=*/false, a, /*neg_b=*/false, b,
                /*c_mod=*/(short)0, acc, /*reuse_a=*/false, /*reuse_b=*/false);
        }
        ap += 32;               // 8 steps * K=4 floats
        bp += 16 * WPITCH;      // 8 steps * 2 k-pair rows
    }

    // ---- epilogue: bias + fast sigmoid (v_exp + v_rcp), scatter rows back
    // C/D layout: VGPR r holds M = r + 8*half, N = n0 + l
    const float bv = bias[c * OUT_SZ + n0 + l];
    const int base = tile * TILE_M;
    #pragma unroll
    for (int r = 0; r < 8; ++r) {
        int m  = r + half * 8;
        int gm = base + m;
        if (gm < count) {
            int s = Si[m];
            float y = acc[r] + bv;
            y = __builtin_amdgcn_rcpf(1.0f + __expf(-y));   // sigmoid, hw rcp
            out[(size_t)s * OUT_SZ + n0 + l] = y;
        }
    }
}

// ---------------------------------------------------------------------------
// Launch: zero counters -> bucket -> grouped WMMA GEMM (all on `stream`)
// ---------------------------------------------------------------------------
extern "C" void kernel_launch(void* const* d_in, const int* in_sizes, int n_in,
                              void* d_out, int out_size, void* d_ws, size_t ws_size,
                              hipStream_t stream) {
    const float* x    = (const float*)d_in[0];
    const int*   cam  = (const int*)  d_in[1];
    const float* W    = (const float*)d_in[2];
    const float* bias = (const float*)d_in[3];
    float* out = (float*)d_out;

    int* counts = (int*)d_ws;          // 16 ints (256B reserved)
    int* idx    = counts + 64;         // 16 x 8192 ints

    fc_zero_counts<<<1, 32, 0, stream>>>(counts);
    fc_bucket<<<BATCH / 256, 256, 0, stream>>>(cam, counts, idx);

    dim3 grid(NUM_CAM, BATCH / TILE_M);   // worst case: all samples on one camera
    fc_expert_gemm<<<grid, 256, 0, stream>>>(x, W, bias, counts, idx, out);
}